// PropertyPredictionMLP_2104533975134
// MI455X (gfx1250) — compile-verified
//
#include <hip/hip_runtime.h>

typedef __attribute__((ext_vector_type(2))) float v2f;
typedef __attribute__((ext_vector_type(8))) float v8f;

__device__ __forceinline__ float silu_f(float x) {
  return x / (1.0f + __expf(-x));
}

// Accumulate C += A(16xK) * B(KxN-tile) using V_WMMA_F32_16X16X4_F32.
// Per-lane fragment layout (ISA 7.12.2, 32-bit operands):
//   A 16x4 : lanes 0-15 -> M=lane, VGPR{0,1}=K{0,1}; lanes 16-31 -> K{2,3}
//   B 4x16 : lanes 0-15 -> N=lane, VGPR{0,1}=K{0,1}; lanes 16-31 -> K{2,3}
// Ap must already point at row M=lane16, element (k=2*khalf); Bp at the
// weight column N=col, element (k=2*khalf). Both advance by 4 per step.
__device__ __forceinline__ v8f wmma_acc(const float* Ap, const float* Bp,
                                        int K, v8f acc) {
#pragma unroll 8
  for (int k = 0; k < K; k += 4) {
    v2f a = *(const v2f*)(Ap + k);
    v2f b = *(const v2f*)(Bp + k);
    acc = __builtin_amdgcn_wmma_f32_16x16x4_f32(false, a, false, b,
                                                (short)0, acc, false, false);
  }
  return acc;
}

__global__ void zero_kernel(float* p, long long n) {
  long long i = (long long)blockIdx.x * blockDim.x + threadIdx.x;
  long long stride = (long long)gridDim.x * blockDim.x;
  for (; i < n; i += stride) p[i] = 0.0f;
}

// 32 threads per edge: float4 load of 4 dims + 4 f32 atomics into node sums.
__global__ __launch_bounds__(256) void edge_kernel(const float* __restrict__ e,
                                                   const int* __restrict__ dst,
                                                   float* esum, float* ecnt,
                                                   int n_edges) {
  int t = blockIdx.x * blockDim.x + threadIdx.x;
  int edge = t >> 5;
  if (edge >= n_edges) return;
  int sub = t & 31;
  int d = dst[edge];
  const float4 ev = *(const float4*)(e + (size_t)edge * 128 + sub * 4);
  float* base = esum + (size_t)d * 128 + sub * 4;
  atomicAdd(base + 0, ev.x);
  atomicAdd(base + 1, ev.y);
  atomicAdd(base + 2, ev.z);
  atomicAdd(base + 3, ev.w);
  if (sub == 0) atomicAdd(&ecnt[d], 1.0f);
}

// One workgroup (8 wave32) per 16-node tile. All GEMMs via f32 WMMA.
__global__ __launch_bounds__(256) void node_kernel(
    const float* __restrict__ s, const float* __restrict__ v,
    const float* __restrict__ We, const float* __restrict__ be,
    const float* __restrict__ Wv0, const float* __restrict__ Ws1w,
    const float* __restrict__ Ws1b, const float* __restrict__ Ws2w,
    const float* __restrict__ Ws2b, const int* __restrict__ batch,
    const float* __restrict__ esum, const float* __restrict__ ecnt,
    float* spool, float* epool, float* bcnt, int n) {
  // +2 padding -> (row stride mod 64) = 2 banks: conflict-free A reads.
  __shared__ float eagg[16][130];
  __shared__ float vdot[16][130];
  __shared__ float hbuf[16][258];
  __shared__ int bat[16];

  const int n0 = blockIdx.x * 16;
  const int wave = threadIdx.x >> 5;
  const int lane = threadIdx.x & 31;
  const int lane16 = lane & 15;
  const int khalf = lane >> 4;   // 0 or 1
  const int kh2 = khalf << 1;    // K offset 0 or 2
  const int mhi = khalf << 3;    // C row offset 0 or 8

  // ---- stage 0: e_agg = 0.5 * esum/max(cnt,1) into LDS; batch ids; bcnt ----
  for (int i = threadIdx.x; i < 16 * 128; i += 256) {
    int r = i >> 7, c = i & 127;
    int node = n0 + r;
    float val = 0.0f;
    if (node < n) val = 0.5f * esum[(size_t)node * 128 + c] / fmaxf(ecnt[node], 1.0f);
    eagg[r][c] = val;
  }
  if (threadIdx.x < 16) {
    int node = n0 + threadIdx.x;
    int b = (node < n) ? batch[node] : 0;
    bat[threadIdx.x] = b;
    if (node < n) atomicAdd(&bcnt[b], 1.0f);
  }
  __syncthreads();

  int mrow = n0 + lane16;
  if (mrow >= n) mrow = n - 1;  // clamp A-row loads for the (nonexistent) tail

  // ---- stage 1: vv = v@Wv0^T per spatial channel; vdot = clip(sum_c vv^2) ----
  // Each wave owns one 16-wide slice of VDIM=128.
  {
    const int nb = wave * 16;
    const int col = nb + lane16;
    v8f sq = {};
    for (int c = 0; c < 3; ++c) {
      v8f acc = {};
      acc = wmma_acc(v + (size_t)mrow * 384 + c * 128 + kh2,
                     Wv0 + (size_t)col * 128 + kh2, 128, acc);
      sq += acc * acc;
    }
#pragma unroll
    for (int r = 0; r < 8; ++r)
      vdot[r + mhi][col] = fmaxf(sq[r], 1e-6f);
  }

  // ---- stage 2: e_feat = silu(e_agg@We^T + be); atomic-add into epool ----
#pragma unroll
  for (int i = 0; i < 2; ++i) {
    const int col = (wave * 2 + i) * 16 + lane16;  // over SDIM=256
    v8f acc = {};
    acc = wmma_acc(&eagg[lane16][kh2], We + (size_t)col * 128 + kh2, 128, acc);
    const float bias = be[col];
#pragma unroll
    for (int r = 0; r < 8; ++r) {
      int m = r + mhi;
      if (n0 + m < n)
        atomicAdd(&epool[(size_t)bat[m] * 256 + col], silu_f(acc[r] + bias));
    }
  }
  __syncthreads();  // vdot ready for stage 3

  // ---- stage 3: h = silu([s | vdot] @ Ws1^T + b), K = 256 + 128 ----
#pragma unroll
  for (int i = 0; i < 2; ++i) {
    const int col = (wave * 2 + i) * 16 + lane16;
    v8f acc = {};
    acc = wmma_acc(s + (size_t)mrow * 256 + kh2,
                   Ws1w + (size_t)col * 384 + kh2, 256, acc);
    acc = wmma_acc(&vdot[lane16][kh2],
                   Ws1w + (size_t)col * 384 + 256 + kh2, 128, acc);
    const float bias = Ws1b[col];
#pragma unroll
    for (int r = 0; r < 8; ++r)
      hbuf[r + mhi][col] = silu_f(acc[r] + bias);
  }
  __syncthreads();

  // ---- stage 4: s_out = h @ Ws2^T + b; atomic-add into spool ----
#pragma unroll
  for (int i = 0; i < 2; ++i) {
    const int col = (wave * 2 + i) * 16 + lane16;
    v8f acc = {};
    acc = wmma_acc(&hbuf[lane16][kh2], Ws2w + (size_t)col * 256 + kh2, 256, acc);
    const float bias = Ws2b[col];
#pragma unroll
    for (int r = 0; r < 8; ++r) {
      int m = r + mhi;
      if (n0 + m < n)
        atomicAdd(&spool[(size_t)bat[m] * 256 + col], acc[r] + bias);
    }
  }
}

// g = (spool+epool)/cnt ; out = silu(g@Pm1^T + b1) @ Pm2^T + b2
__global__ __launch_bounds__(256) void final_kernel(
    const float* __restrict__ spool, const float* __restrict__ epool,
    const float* __restrict__ bcnt, const float* __restrict__ Pm1w,
    const float* __restrict__ Pm1b, const float* __restrict__ Pm2w,
    const float* __restrict__ Pm2b, float* out) {
  __shared__ float g[256];
  __shared__ float red[256];
  const int b = blockIdx.x, t = threadIdx.x;
  const float inv = 1.0f / fmaxf(bcnt[b], 1.0f);
  g[t] = (spool[(size_t)b * 256 + t] + epool[(size_t)b * 256 + t]) * inv;
  __syncthreads();
  float acc = Pm1b[t];
#pragma unroll 8
  for (int c = 0; c < 256; ++c) acc += g[c] * Pm1w[(size_t)t * 256 + c];
  red[t] = silu_f(acc) * Pm2w[t];
  __syncthreads();
  for (int off = 128; off > 0; off >>= 1) {
    if (t < off) red[t] += red[t + off];
    __syncthreads();
  }
  if (t == 0) out[b] = red[0] + Pm2b[0];
}

extern "C" void kernel_launch(void* const* d_in, const int* in_sizes, int n_in,
                              void* d_out, int out_size, void* d_ws, size_t ws_size,
                              hipStream_t stream) {
  const float* s    = (const float*)d_in[0];
  const float* v    = (const float*)d_in[1];
  // d_in[2] = p : contributes exactly 0 (sum(p)*0.0), skipped
  const float* e    = (const float*)d_in[3];
  const float* We   = (const float*)d_in[4];
  const float* be   = (const float*)d_in[5];
  const float* Wv0  = (const float*)d_in[6];
  const float* Ws1w = (const float*)d_in[7];
  const float* Ws1b = (const float*)d_in[8];
  const float* Ws2w = (const float*)d_in[9];
  const float* Ws2b = (const float*)d_in[10];
  const float* Pm1w = (const float*)d_in[11];
  const float* Pm1b = (const float*)d_in[12];
  const float* Pm2w = (const float*)d_in[13];
  const float* Pm2b = (const float*)d_in[14];
  const int* batch  = (const int*)d_in[15];
  const int* eidx   = (const int*)d_in[16];

  const int n       = in_sizes[0] / 256;   // N_NODES
  const int n_edges = in_sizes[3] / 128;   // N_EDGES
  const int bs      = out_size;            // output is [bs, 1]

  // workspace layout (f32)
  float* esum  = (float*)d_ws;                    // n * 128
  float* ecnt  = esum + (size_t)n * 128;          // n
  float* spool = ecnt + n;                        // bs * 256
  float* epool = spool + (size_t)bs * 256;        // bs * 256
  float* bcnt  = epool + (size_t)bs * 256;        // bs
  const long long zn = (long long)n * 128 + n + (long long)bs * 512 + bs;

  zero_kernel<<<2048, 256, 0, stream>>>(esum, zn);

  {
    long long tot = (long long)n_edges * 32;
    int blocks = (int)((tot + 255) / 256);
    edge_kernel<<<blocks, 256, 0, stream>>>(e, eidx + n_edges, esum, ecnt, n_edges);
  }

  node_kernel<<<(n + 15) / 16, 256, 0, stream>>>(
      s, v, We, be, Wv0, Ws1w, Ws1b, Ws2w, Ws2b, batch, esum, ecnt,
      spool, epool, bcnt, n);

  final_kernel<<<bs, 256, 0, stream>>>(spool, epool, bcnt, Pm1w, Pm1b,
                                       Pm2w, Pm2b, (float*)d_out);
}